// E72MatrixSelfGateCell_87256555586126
// MI455X (gfx1250) — compile-verified
//
#include <hip/hip_runtime.h>
#include <math.h>

#define T_STEPS 2048
#define BATCH   8
#define DIMK    1024
#define NST     64

typedef __attribute__((ext_vector_type(2))) float v2f;
typedef __attribute__((ext_vector_type(8))) float v8f;

__device__ __forceinline__ float sigmoidf_(float x) {
    return 1.0f / (1.0f + __expf(-x));
}

// ---------------------------------------------------------------------------
// Kernel 1: fused 4-way projection GEMM via V_WMMA_F32_16X16X4_F32.
// out[tb, n] = sum_d x[tb, d] * W[n, d];  alpha gets bias + sigmoid fused.
// One wave computes a 16-row (M) strip of ONE weight matrix (all 64 cols =
// 4 accumulators). 256-thread blocks = 8 waves. Tasks = 1024 Mtiles * 4 W.
// ---------------------------------------------------------------------------
__global__ void __launch_bounds__(256)
proj_wmma_kernel(const float* __restrict__ x,
                 const float* __restrict__ Wk, const float* __restrict__ Wv,
                 const float* __restrict__ Wq, const float* __restrict__ Wa,
                 const float* __restrict__ b_alpha,
                 float* __restrict__ k_all, float* __restrict__ v_all,
                 float* __restrict__ q_all, float* __restrict__ a_all)
{
    const int wave = threadIdx.x >> 5;
    const int lane = threadIdx.x & 31;
    const int task = blockIdx.x * 8 + wave;     // 0..4095
    const int mt   = task >> 2;                 // 0..1023  (16-row strip)
    const int wsel = task & 3;                  // which weight matrix

    const float* W =
        (wsel == 0) ? Wk : (wsel == 1) ? Wv : (wsel == 2) ? Wq : Wa;
    float* outp =
        (wsel == 0) ? k_all : (wsel == 1) ? v_all : (wsel == 2) ? q_all : a_all;

    const int m  = lane & 15;                   // M index (also N index for B/C)
    const int hi = lane >> 4;                   // selects K-pair {0,1} vs {2,3}

    // A fragment source: row (mt*16+m) of x, contiguous float2 at kk+2*hi
    const float* arow = x + (size_t)(mt * 16 + m) * DIMK + 2 * hi;

    // B fragment sources: rows (n0+n) of W, n = lane&15, 4 column tiles
    const float* brow0 = W + (size_t)( 0 + m) * DIMK + 2 * hi;
    const float* brow1 = W + (size_t)(16 + m) * DIMK + 2 * hi;
    const float* brow2 = W + (size_t)(32 + m) * DIMK + 2 * hi;
    const float* brow3 = W + (size_t)(48 + m) * DIMK + 2 * hi;

    v8f acc0 = {}, acc1 = {}, acc2 = {}, acc3 = {};

    #pragma unroll 4
    for (int kk = 0; kk < DIMK; kk += 4) {
        v2f a  = *(const v2f*)(arow  + kk);
        v2f b0 = *(const v2f*)(brow0 + kk);
        v2f b1 = *(const v2f*)(brow1 + kk);
        v2f b2 = *(const v2f*)(brow2 + kk);
        v2f b3 = *(const v2f*)(brow3 + kk);
        acc0 = __builtin_amdgcn_wmma_f32_16x16x4_f32(false, a, false, b0,
                                                     (short)0, acc0, false, false);
        acc1 = __builtin_amdgcn_wmma_f32_16x16x4_f32(false, a, false, b1,
                                                     (short)0, acc1, false, false);
        acc2 = __builtin_amdgcn_wmma_f32_16x16x4_f32(false, a, false, b2,
                                                     (short)0, acc2, false, false);
        acc3 = __builtin_amdgcn_wmma_f32_16x16x4_f32(false, a, false, b3,
                                                     (short)0, acc3, false, false);
    }

    // Epilogue. C layout: lane L holds (m = r + 8*(L>>4), n = L&15) in acc[r].
    const bool is_alpha = (wsel == 3);
    #pragma unroll
    for (int nt = 0; nt < 4; ++nt) {
        v8f acc = (nt == 0) ? acc0 : (nt == 1) ? acc1 : (nt == 2) ? acc2 : acc3;
        const int ng = nt * 16 + m;             // global column 0..63
        const float bias = is_alpha ? b_alpha[ng] : 0.0f;
        #pragma unroll
        for (int r = 0; r < 8; ++r) {
            const int mrow = mt * 16 + r + hi * 8;
            float v = acc[r];
            if (is_alpha) v = sigmoidf_(v + bias);
            outp[(size_t)mrow * NST + ng] = v;
        }
    }
}

// ---------------------------------------------------------------------------
// Kernel 2: sequential scan. One block per batch, 64 threads; thread i owns
// row i of S (64 VGPRs). Restructured output:
//   out_i = alpha_i * (S_old q)_i + (1-alpha_i) * vg_i * (k . q)
// so S@k, S@q, k.q fuse into one pass over S_old, then rank-1 update.
// ---------------------------------------------------------------------------
__global__ void __launch_bounds__(64)
scan_kernel(const float* __restrict__ k_all, const float* __restrict__ v_all,
            const float* __restrict__ q_all, const float* __restrict__ a_all,
            const float* __restrict__ S0,   const float* __restrict__ d_g,
            const float* __restrict__ b_g,
            float* __restrict__ out,        // [T,B,N]
            float* __restrict__ S_final)    // [B,N,N]
{
    const int b = blockIdx.x;
    const int i = threadIdx.x;              // row index 0..63

    __shared__ float k_s[NST];
    __shared__ float q_s[NST];

    float S_row[NST];
    #pragma unroll
    for (int j = 0; j < NST; ++j)
        S_row[j] = S0[(size_t)b * NST * NST + (size_t)i * NST + j];

    const float dg = d_g[i];
    const float bg = b_g[i];

    for (int t = 0; t < T_STEPS; ++t) {
        const size_t base = ((size_t)t * BATCH + b) * NST;
        const float ki = k_all[base + i];
        const float vi = v_all[base + i];
        const float qi = q_all[base + i];
        const float ai = a_all[base + i];

        __syncthreads();                     // prior iteration done with LDS
        k_s[i] = ki;
        q_s[i] = qi;
        __syncthreads();

        float r = 0.0f, sq = 0.0f, kq = 0.0f;
        #pragma unroll
        for (int j = 0; j < NST; ++j) {
            const float kj = k_s[j];
            const float qj = q_s[j];
            r  = fmaf(S_row[j], kj, r);
            sq = fmaf(S_row[j], qj, sq);
            kq = fmaf(kj, qj, kq);
        }

        const float g  = sigmoidf_(fmaf(dg, r, bg));
        const float vg = vi * g;
        const float om = 1.0f - ai;
        const float c  = om * vg;

        #pragma unroll
        for (int j = 0; j < NST; ++j)
            S_row[j] = fmaf(c, k_s[j], ai * S_row[j]);

        const float o = fmaf(ai, sq, c * kq);
        out[base + i] = o * o * sigmoidf_(o);   // out * silu(out)
    }

    #pragma unroll
    for (int j = 0; j < NST; ++j)
        S_final[(size_t)b * NST * NST + (size_t)i * NST + j] = S_row[j];
}

// ---------------------------------------------------------------------------
extern "C" void kernel_launch(void* const* d_in, const int* in_sizes, int n_in,
                              void* d_out, int out_size, void* d_ws, size_t ws_size,
                              hipStream_t stream)
{
    (void)in_sizes; (void)n_in; (void)out_size; (void)ws_size;

    const float* x       = (const float*)d_in[0];
    const float* S0      = (const float*)d_in[1];
    const float* Wk      = (const float*)d_in[2];
    const float* Wv      = (const float*)d_in[3];
    const float* Wq      = (const float*)d_in[4];
    const float* Wa      = (const float*)d_in[5];
    const float* b_alpha = (const float*)d_in[6];
    const float* d_g     = (const float*)d_in[7];
    const float* b_g     = (const float*)d_in[8];

    float* out = (float*)d_out;                              // [T,B,N]
    float* Sf  = out + (size_t)T_STEPS * BATCH * NST;        // [B,N,N]

    // Workspace: 4 projection buffers of T*B*N floats (16 MB total).
    float* ws     = (float*)d_ws;
    const size_t per = (size_t)T_STEPS * BATCH * NST;
    float* k_all = ws;
    float* v_all = ws + per;
    float* q_all = ws + 2 * per;
    float* a_all = ws + 3 * per;

    // 1024 M-tiles * 4 weights = 4096 wave tasks, 8 waves per 256-thread block
    proj_wmma_kernel<<<512, 256, 0, stream>>>(x, Wk, Wv, Wq, Wa, b_alpha,
                                              k_all, v_all, q_all, a_all);

    scan_kernel<<<BATCH, 64, 0, stream>>>(k_all, v_all, q_all, a_all,
                                          S0, d_g, b_g, out, Sf);
}